// EglAPPNPConv_56599079026983
// MI455X (gfx1250) — compile-verified
//
#include <hip/hip_runtime.h>
#include <hip/hip_bf16.h>
#include <stdint.h>

#define D_FEAT 64
#define ALPHA_F 0.1f
#define K_ITERS 10

#define EDGES_PER_STEP 16            // 256 threads / 16 lanes-per-edge
#define STEPS_PER_BLOCK 4
#define EDGES_PER_BLOCK (EDGES_PER_STEP * STEPS_PER_BLOCK)   // 64

// ---------------- CDNA5 async global->LDS staging ----------------
#if __has_builtin(__builtin_amdgcn_s_wait_asynccnt)
#define WAIT_ASYNC(n) __builtin_amdgcn_s_wait_asynccnt(n)
#else
#define WAIT_ASYNC(n) asm volatile("s_wait_asynccnt %0" ::"i"(n) : "memory")
#endif

#if __has_builtin(__builtin_amdgcn_global_load_async_to_lds_b128)
typedef int v4i_t __attribute__((vector_size(16)));
typedef __attribute__((address_space(1))) v4i_t* g_v4i_p;   // global v4i*
typedef __attribute__((address_space(3))) v4i_t* l_v4i_p;   // LDS v4i*
#endif

__device__ __forceinline__ void async_gather16(const float* gsrc, float* lds_dst) {
#if __has_builtin(__builtin_amdgcn_global_load_async_to_lds_b128)
  __builtin_amdgcn_global_load_async_to_lds_b128(
      (g_v4i_p)(uintptr_t)(const void*)gsrc,
      (l_v4i_p)(uint32_t)(uintptr_t)(void*)lds_dst,
      0, 0);
#else
  // Fallback: stage through registers (still correct; no ASYNCcnt traffic).
  float4 v = *(const float4*)gsrc;
  *(float4*)lds_dst = v;
#endif
}

__device__ __forceinline__ void atomic_add_f32_agent(float* p, float v) {
  __hip_atomic_fetch_add(p, v, __ATOMIC_RELAXED, __HIP_MEMORY_SCOPE_AGENT);
}

// ---------------- kernels ----------------

__global__ void zero_ws_kernel(int* __restrict__ deg, float4* __restrict__ agg4,
                               int n_nodes, int n_agg4) {
  int i = blockIdx.x * blockDim.x + threadIdx.x;
  if (i < n_nodes) deg[i] = 0;
  if (i < n_agg4) agg4[i] = make_float4(0.f, 0.f, 0.f, 0.f);
}

__global__ void deg_kernel(const int* __restrict__ ed, int* __restrict__ deg,
                           int n_edges) {
  int i = blockIdx.x * blockDim.x + threadIdx.x;
  if (i < n_edges) {
    __builtin_prefetch(ed + i + 4096, 0, 1);   // stream edge list ahead
    atomicAdd(&deg[ed[i]], 1);
  }
}

__global__ void norm_kernel(const int* __restrict__ deg, float* __restrict__ norm,
                            int n_nodes) {
  int i = blockIdx.x * blockDim.x + threadIdx.x;
  if (i < n_nodes) {
    float d = (float)deg[i];
    norm[i] = rsqrtf(fmaxf(d, 1.0f));
  }
}

// Scatter phase: msg = h[src]*norm[src], atomically accumulated into agg[dst].
// 16 lanes per edge (float4 each) -> one row = 2 coalesced 128B lines.
// Rows are staged via async global->LDS loads, double-buffered over 4 steps.
__global__ __launch_bounds__(256) void scatter_kernel(
    const float* __restrict__ h, const float* __restrict__ norm,
    const int* __restrict__ es, const int* __restrict__ ed,
    float* __restrict__ agg, int n_edges) {
  __shared__ float buf[2][EDGES_PER_STEP * D_FEAT];   // 2 x 4KB staging
  const int le = threadIdx.x >> 4;                    // local edge 0..15
  const int c  = threadIdx.x & 15;                    // float4 chunk 0..15
  const int base = blockIdx.x * EDGES_PER_BLOCK;

  float* myLds[2] = { &buf[0][le * D_FEAT + c * 4],
                      &buf[1][le * D_FEAT + c * 4] };
  int srcs[STEPS_PER_BLOCK];
  int dsts[STEPS_PER_BLOCK];

  auto issue = [&](int s) {
    int e = base + s * EDGES_PER_STEP + le;
    e = (e < n_edges) ? e : (n_edges - 1);            // clamp: uniform issue count
    int src = es[e];
    srcs[s] = src;
    dsts[s] = ed[e];
    async_gather16(h + (size_t)src * D_FEAT + c * 4, myLds[s & 1]);
  };

  __builtin_prefetch(es + base + EDGES_PER_BLOCK * 32, 0, 1);
  __builtin_prefetch(ed + base + EDGES_PER_BLOCK * 32, 0, 1);

  issue(0);
#pragma unroll
  for (int s = 0; s < STEPS_PER_BLOCK; ++s) {
    if (s + 1 < STEPS_PER_BLOCK) {
      issue(s + 1);      // outstanding: step s and s+1
      WAIT_ASYNC(1);     // step s landed in LDS
    } else {
      WAIT_ASYNC(0);     // last step landed
    }
    asm volatile("" ::: "memory");                    // keep LDS reads below the wait
    int e = base + s * EDGES_PER_STEP + le;
    if (e < n_edges) {
      float4 v = *(const float4*)myLds[s & 1];
      float sc = norm[srcs[s]];
      float* a = agg + (size_t)dsts[s] * D_FEAT + c * 4;
      atomic_add_f32_agent(a + 0, v.x * sc);
      atomic_add_f32_agent(a + 1, v.y * sc);
      atomic_add_f32_agent(a + 2, v.z * sc);
      atomic_add_f32_agent(a + 3, v.w * sc);
    }
  }
}

// Node update: h = (1-alpha)*agg*norm + alpha*feat0, and re-zero agg for the
// next iteration (fused, saving a full memset pass over 25.6 MB).
__global__ void update_kernel(const float* __restrict__ feat,
                              const float* __restrict__ norm,
                              float* __restrict__ agg, float* __restrict__ h,
                              int n_nodes) {
  int t = blockIdx.x * blockDim.x + threadIdx.x;
  int total = n_nodes * (D_FEAT / 4);
  if (t < total) {
    int node = t >> 4;
    float nv = norm[node];
    float4* a4 = (float4*)agg + t;
    float4 av = *a4;
    float4 fv = ((const float4*)feat)[t];
    float4 r;
    r.x = (1.0f - ALPHA_F) * (av.x * nv) + ALPHA_F * fv.x;
    r.y = (1.0f - ALPHA_F) * (av.y * nv) + ALPHA_F * fv.y;
    r.z = (1.0f - ALPHA_F) * (av.z * nv) + ALPHA_F * fv.z;
    r.w = (1.0f - ALPHA_F) * (av.w * nv) + ALPHA_F * fv.w;
    ((float4*)h)[t] = r;
    *a4 = make_float4(0.f, 0.f, 0.f, 0.f);
  }
}

// ---------------- host launcher ----------------

static inline size_t align_up(size_t x, size_t a) { return (x + a - 1) & ~(a - 1); }

extern "C" void kernel_launch(void* const* d_in, const int* in_sizes, int n_in,
                              void* d_out, int out_size, void* d_ws, size_t ws_size,
                              hipStream_t stream) {
  const float* feat = (const float*)d_in[0];
  const int*   es   = (const int*)d_in[1];
  const int*   ed   = (const int*)d_in[2];
  float*       h    = (float*)d_out;

  const int n_nodes = in_sizes[0] / D_FEAT;
  const int n_edges = in_sizes[1];

  char* ws = (char*)d_ws;
  size_t off = 0;
  float* norm = (float*)(ws + off); off = align_up(off + (size_t)n_nodes * 4, 256);
  int*   deg  = (int*)(ws + off);   off = align_up(off + (size_t)n_nodes * 4, 256);
  float* agg  = (float*)(ws + off);
  const int n_agg4 = n_nodes * (D_FEAT / 4);

  // 1) zero degree counters + agg accumulator (ws is poisoned by harness)
  {
    int total = (n_agg4 > n_nodes) ? n_agg4 : n_nodes;
    zero_ws_kernel<<<(total + 255) / 256, 256, 0, stream>>>(deg, (float4*)agg,
                                                            n_nodes, n_agg4);
  }
  // 2) in-degree histogram, 3) norm = rsqrt(max(deg,1))
  deg_kernel<<<(n_edges + 255) / 256, 256, 0, stream>>>(ed, deg, n_edges);
  norm_kernel<<<(n_nodes + 255) / 256, 256, 0, stream>>>(deg, norm, n_nodes);

  // 4) h = feat0
  (void)hipMemcpyAsync(h, feat, (size_t)n_nodes * D_FEAT * sizeof(float),
                       hipMemcpyDeviceToDevice, stream);

  // 5) K propagation rounds (whole working set is L2-resident: 85 MB < 192 MB)
  const int scatter_blocks = (n_edges + EDGES_PER_BLOCK - 1) / EDGES_PER_BLOCK;
  const int update_blocks  = (n_agg4 + 255) / 256;
  for (int k = 0; k < K_ITERS; ++k) {
    scatter_kernel<<<scatter_blocks, 256, 0, stream>>>(h, norm, es, ed, agg,
                                                       n_edges);
    update_kernel<<<update_blocks, 256, 0, stream>>>(feat, norm, agg, h,
                                                     n_nodes);
  }
}